// TrueMultiLayerLattice_16810501996613
// MI455X (gfx1250) — compile-verified
//
#include <hip/hip_runtime.h>
#include <math.h>

// Problem constants (match the reference)
#define DIM   2048
#define TWO_D 4096
#define SEQ   8192
#define BATCH 8
#define LN_EPS 1e-5f

// WMMA vector types (CDNA5 / gfx1250, wave32)
typedef __attribute__((ext_vector_type(2))) float v2f;
typedef __attribute__((ext_vector_type(8))) float v8f;

// Spine steps: (pz, py, px) for k = 3..9 of rel = [0,2,4,12,36,104,304,888,2592,7568]
static const int PZ[7] = {12, 36, 104, 304, 888, 2592, 7568};
static const int PY[7] = { 4, 12,  36, 104, 304,  888, 2592};
static const int PX[7] = { 2,  4,  12,  36, 104,  304,  888};

// ---------------------------------------------------------------------------
// 1) Streaming copy x -> h (512 MB each way; the bandwidth-dominant part)
// ---------------------------------------------------------------------------
__global__ void __launch_bounds__(256) copy_kernel(const float4* __restrict__ src,
                                                   float4* __restrict__ dst,
                                                   long n4) {
  long i = (long)blockIdx.x * blockDim.x + threadIdx.x;
  long stride = (long)gridDim.x * blockDim.x;
  for (; i < n4; i += stride) dst[i] = src[i];
}

// ---------------------------------------------------------------------------
// 2) Per-step prep: A[16][4096] = [v | z] for rows 0..7 (batches), zeros 8..15
// ---------------------------------------------------------------------------
__global__ void __launch_bounds__(256) prep_kernel(
    const float* __restrict__ h, float* __restrict__ A,
    const float* __restrict__ axz, const float* __restrict__ bxy, const float* __restrict__ gx,
    const float* __restrict__ awy, const float* __restrict__ bwx, const float* __restrict__ gw,
    const float* __restrict__ axv, const float* __restrict__ bwv, const float* __restrict__ gv,
    int pz, int py, int px) {
  const int row = blockIdx.x;              // 0..15 (A row)
  if (row >= BATCH) {                      // zero the M-padding rows
    for (int d = threadIdx.x; d < TWO_D; d += 256) A[row * TWO_D + d] = 0.0f;
    return;
  }
  const float* hb = h + (size_t)row * SEQ * DIM;
  for (int d = threadIdx.x; d < DIM; d += 256) {
    float z  = hb[(size_t)pz * DIM + d];
    float y  = hb[(size_t)py * DIM + d];
    float xp = hb[(size_t)px * DIM + d];
    float xn = axz[d] * z + bxy[d] * y + gx[d];
    float w  = awy[d] * y + bwx[d] * xp + gw[d];
    float v  = axv[d] * xn + bwv[d] * w + gv[d];
    A[row * TWO_D + d]       = v;   // first half: v
    A[row * TWO_D + DIM + d] = z;   // second half: z
  }
}

// ---------------------------------------------------------------------------
// 3) Gate GEMM + sigmoid + gated blend, via V_WMMA_F32_16X16X4_F32
//    g = sigmoid(A(16x4096) x gate_w^T(4096x2048) + b); U = g*v + (1-g)*z
//    Grid: 16 blocks x 256 threads (8 waves); each wave owns one 16x16 N-tile.
// ---------------------------------------------------------------------------
#define KCHUNK 512   // 16 x 512 f32 = 32 KB LDS per chunk

__global__ void __launch_bounds__(256) gate_gemm_kernel(
    const float* __restrict__ A,        // [16][4096]
    const float* __restrict__ gate_w,   // [2048][4096] row-major (n, k)
    const float* __restrict__ gate_b,   // [2048]
    float* __restrict__ U) {            // [8][2048]
  __shared__ float ldsA[16 * KCHUNK];

  const int tid  = threadIdx.x;
  const int wave = tid >> 5;             // 0..7 (wave32)
  const int lane = tid & 31;
  const int tile = blockIdx.x * 8 + wave;  // 0..127 N-tiles
  const int nbase = tile * 16;
  const int m16  = lane & 15;            // M (for A frag) / N (for B frag) index
  const int half = lane >> 4;            // 0: K pair {0,1}, 1: K pair {2,3}

  v8f acc = {};  // 16x16 f32 accumulator (8 VGPRs)

  for (int c = 0; c < TWO_D / KCHUNK; ++c) {
    // Cooperative, coalesced stage of the A K-chunk into LDS (float4)
    for (int i = tid; i < 16 * KCHUNK / 4; i += 256) {
      int lin = i * 4;
      int r   = lin / KCHUNK;
      int kk  = lin % KCHUNK;
      ((float4*)ldsA)[i] = *(const float4*)(A + (size_t)r * TWO_D + c * KCHUNK + kk);
    }
    __syncthreads();

    // B rows stream contiguously in K per lane; gate_w is L2-resident (32 MB < 192 MB)
    const float* wrow = gate_w + (size_t)(nbase + m16) * TWO_D + c * KCHUNK;

#pragma unroll 4
    for (int kk = 0; kk < KCHUNK; kk += 4) {
      const int ka = kk + 2 * half;
      v2f a, b;
      // A frag: lane holds A[m16][ka], A[m16][ka+1]  (16x4 MxK layout)
      a.x = ldsA[m16 * KCHUNK + ka];
      a.y = ldsA[m16 * KCHUNK + ka + 1];
      // B frag: lane holds B[ka][n] = gate_w[n][ka], B[ka+1][n]
      b.x = wrow[ka];
      b.y = wrow[ka + 1];
      acc = __builtin_amdgcn_wmma_f32_16x16x4_f32(
          /*neg_a=*/false, a, /*neg_b=*/false, b,
          /*c_mod=*/(short)0, acc, /*reuse_a=*/false, /*reuse_b=*/false);
    }
    __syncthreads();
  }

  // Epilogue. C layout: VGPR i, lanes 0-15 -> (M=i, N=lane); lanes 16-31 -> M=8+i
  // (padding rows). Only half==0 lanes carry real batches.
  if (half == 0) {
    const int n = nbase + m16;
    const float bias = gate_b[n];
#pragma unroll
    for (int i = 0; i < 8; ++i) {
      float t = acc[i] + bias;
      float g = 1.0f / (1.0f + __expf(-t));
      float v = A[(size_t)i * TWO_D + n];          // v part of row i
      float z = A[(size_t)i * TWO_D + DIM + n];    // z part of row i
      U[i * DIM + n] = g * v + (1.0f - g) * z;
    }
  }
}

// ---------------------------------------------------------------------------
// 4) LayerNorm of U row (population var, matching jnp.var) -> h[:, pz, :]
// ---------------------------------------------------------------------------
__global__ void __launch_bounds__(256) ln_kernel(
    const float* __restrict__ U, const float* __restrict__ lnw,
    const float* __restrict__ lnb, float* __restrict__ h, int pz) {
  __shared__ float s_sum[256];
  __shared__ float s_sq[256];
  const int b = blockIdx.x;
  const float* u = U + b * DIM;

  float vals[8];
  float sum = 0.0f, sq = 0.0f;
#pragma unroll
  for (int j = 0; j < 8; ++j) {
    float t = u[threadIdx.x + j * 256];
    vals[j] = t;
    sum += t;
    sq  += t * t;
  }
  s_sum[threadIdx.x] = sum;
  s_sq[threadIdx.x]  = sq;
  __syncthreads();
  for (int off = 128; off > 0; off >>= 1) {
    if (threadIdx.x < off) {
      s_sum[threadIdx.x] += s_sum[threadIdx.x + off];
      s_sq[threadIdx.x]  += s_sq[threadIdx.x + off];
    }
    __syncthreads();
  }
  const float mean = s_sum[0] * (1.0f / DIM);
  const float var  = s_sq[0] * (1.0f / DIM) - mean * mean;
  const float inv  = rsqrtf(var + LN_EPS);

  float* out = h + (size_t)b * SEQ * DIM + (size_t)pz * DIM;
#pragma unroll
  for (int j = 0; j < 8; ++j) {
    int d = threadIdx.x + j * 256;
    out[d] = (vals[j] - mean) * inv * lnw[d] + lnb[d];
  }
}

// ---------------------------------------------------------------------------
extern "C" void kernel_launch(void* const* d_in, const int* in_sizes, int n_in,
                              void* d_out, int out_size, void* d_ws, size_t ws_size,
                              hipStream_t stream) {
  const float* x    = (const float*)d_in[0];
  const float* axz  = (const float*)d_in[1];
  const float* bxy  = (const float*)d_in[2];
  const float* gx   = (const float*)d_in[3];
  const float* awy  = (const float*)d_in[4];
  const float* bwx  = (const float*)d_in[5];
  const float* gw   = (const float*)d_in[6];
  const float* axv  = (const float*)d_in[7];
  const float* bwv  = (const float*)d_in[8];
  const float* gv   = (const float*)d_in[9];
  const float* gwt  = (const float*)d_in[10];  // gate_w [2048][4096]
  const float* gb   = (const float*)d_in[11];  // gate_b
  const float* lnw  = (const float*)d_in[12];
  const float* lnb  = (const float*)d_in[13];

  float* h = (float*)d_out;

  // Workspace layout: A[16][4096] then U[8][2048]
  float* A = (float*)d_ws;
  float* U = A + 16 * TWO_D;

  // h = x (bandwidth-dominant stream copy)
  const long n4 = (long)BATCH * SEQ * DIM / 4;
  copy_kernel<<<8192, 256, 0, stream>>>((const float4*)x, (float4*)h, n4);

  // 7 sequential spine steps
  for (int s = 0; s < 7; ++s) {
    prep_kernel<<<16, 256, 0, stream>>>(h, A, axz, bxy, gx, awy, bwx, gw,
                                        axv, bwv, gv, PZ[s], PY[s], PX[s]);
    gate_gemm_kernel<<<16, 256, 0, stream>>>(A, gwt, gb, U);
    ln_kernel<<<8, 256, 0, stream>>>(U, lnw, lnb, h, PZ[s]);
  }
}